// CASSI_layer_87497073755004
// MI455X (gfx1250) — compile-verified
//
#include <hip/hip_runtime.h>

// CASSI layer for MI455X (gfx1250, wave32).
// Memory-bound (~150MB traffic vs ~0.55 GFLOP) -> keep fp32, use
// V_WMMA_F32_16X16X4_F32 for both banded GEMM phases.
// All band/tail predication is branch-free: clamped load addresses +
// cndmask on data, and cndmask'd *store addresses* (dump row / dump word)
// so prologue, K-loop and epilogue are straight-line code.

typedef float v2f __attribute__((ext_vector_type(2)));
typedef float v8f __attribute__((ext_vector_type(8)));

#define B_   4
#define M_   256
#define L_   24
#define S_   22
#define SP_  24    // Yt rows padded: 22 real + dump row 22 (+1 align)
#define W_   279   // M + L - 1
#define WP_  288   // W padded to multiple of 16
#define NTW  18    // w tiles (288/16)
#define NTN  16    // n tiles (256/16)

__device__ __forceinline__ unsigned map_f32(float x) {
  // order-preserving float -> uint mapping (all finite values map > 0)
  unsigned u = __float_as_uint(x);
  return (u & 0x80000000u) ? ~u : (u | 0x80000000u);
}

__device__ __forceinline__ int clampi(int v, int lo, int hi) {
  return v < lo ? lo : (v > hi ? hi : v);
}

// ---------------------------------------------------------------------------
// Phase 1: Yt[b][m][s][w] = sum_n H2[m,n,s] * x[b,m,n,w-n]   (0 <= w-n < L)
// One wave per (b, m, w-tile). GEMM: D(16w x 16s) += A(16w x 4n) * B(4n x 16s)
// Also accumulates sum(Y^2) over the valid (w<279, s<22) region.
// ---------------------------------------------------------------------------
__global__ __launch_bounds__(128) void cassi_phase1(
    const float* __restrict__ x, const float* __restrict__ H,
    float* __restrict__ Yt, float* __restrict__ sumYY)
{
  const int lane = threadIdx.x & 31;
  const int wave = blockIdx.x * 4 + (threadIdx.x >> 5);
  const int wt = wave % NTW;
  const int bm = wave / NTW;
  const int m  = bm & (M_ - 1);
  const int b  = bm >> 8;

  const int lo16 = lane & 15;
  const int hi   = lane >> 4;
  const int w0   = wt * 16;
  const int w_me = w0 + lo16;                       // A-fragment row (M dim)

  const float* xb = x + (size_t)(b * M_ + m) * M_ * L_;   // x[b][m][n][i]
  const float* Hm = H + (size_t)m * M_ * S_;              // H2[m][n][s]

  v8f c0 = {};   // N-tile s0 = 0
  v8f c1 = {};   // N-tile s0 = 16

  int nstart = w0 - (L_ - 1); if (nstart < 0) nstart = 0;
  nstart &= ~3;
  int nend = w0 + 15; if (nend > M_ - 1) nend = M_ - 1;

  const bool s1ok = (16 + lo16 < S_);
  const int  s1c  = s1ok ? 16 + lo16 : S_ - 1;      // clamped load col
  const int  srow1 = s1ok ? 16 + lo16 : S_;         // store row (22 = dump)

  for (int n0 = nstart; n0 <= nend; n0 += 4) {
    v2f a, b0v, b1v;
#pragma unroll
    for (int j = 0; j < 2; ++j) {
      int k = j + 2 * hi;                 // f32 frag layout: VGPR j, half hi
      int n = n0 + k;
      int i = w_me - n;
      bool aok = (n < M_) & ((unsigned)i < (unsigned)L_);
      bool nok = (n < M_);
      int nc = n < M_ ? n : M_ - 1;
      int ic = clampi(i, 0, L_ - 1);
      // unconditional in-bounds loads, then select-to-zero
      float av = xb[nc * L_ + ic];
      float h0 = Hm[nc * S_ + lo16];
      float h1 = Hm[nc * S_ + s1c];
      a[j]   = aok ? av : 0.0f;
      b0v[j] = nok ? h0 : 0.0f;
      b1v[j] = (nok & s1ok) ? h1 : 0.0f;
    }
    c0 = __builtin_amdgcn_wmma_f32_16x16x4_f32(false, a, false, b0v,
                                               (short)0, c0, false, false);
    c1 = __builtin_amdgcn_wmma_f32_16x16x4_f32(false, a, false, b1v,
                                               (short)0, c1, false, false);
  }

  // Straight-line store of transposed Yt[s][w]; invalid cols hit dump row 22,
  // pads w in [279,288) hold benign junk. Sum Y^2 over the valid region.
  float* Ybm = Yt + (size_t)(b * M_ + m) * SP_ * WP_;
  float* r0 = Ybm + (size_t)lo16 * WP_ + w0 + 8 * hi;
  float* r1 = Ybm + (size_t)srow1 * WP_ + w0 + 8 * hi;
  float sq = 0.0f;
#pragma unroll
  for (int r = 0; r < 8; ++r) {
    int w = w0 + r + 8 * hi;          // D row
    bool wok = (w < W_);
    float v0 = c0[r];
    float v1 = c1[r];
    r0[r] = v0;
    r1[r] = v1;
    sq += wok ? v0 * v0 : 0.0f;
    sq += (wok & s1ok) ? v1 * v1 : 0.0f;
  }
#pragma unroll
  for (int off = 16; off > 0; off >>= 1) sq += __shfl_down(sq, off, 32);
  if (lane == 0) atomicAdd(sumYY, sq);
}

// ---------------------------------------------------------------------------
// Phase 2: Z = H2m(256x22) * Yt(22x288) banded; X[n,l] = Z[n,n+l] + noise.
// One wave per (b, m, n-tile); A fragment reused across 3 banded u-tiles.
// noise term: sqrt(sigm) * eps[b,m,n+l] * sum_s H2[m,n,s].
// ---------------------------------------------------------------------------
__global__ __launch_bounds__(128) void cassi_phase2(
    const float* __restrict__ H, const float* __restrict__ eps,
    const float* __restrict__ Yt, const float* __restrict__ sumYY,
    float* __restrict__ Xout, unsigned* __restrict__ umax,
    float* __restrict__ dump)
{
  const int lane = threadIdx.x & 31;
  const int wave = blockIdx.x * 4 + (threadIdx.x >> 5);
  const int t  = wave % NTN;                 // n tile
  const int bm = wave / NTN;
  const int m  = bm & (M_ - 1);
  const int b  = bm >> 8;

  const int lo16 = lane & 15;
  const int hi   = lane >> 4;
  const int n0   = t * 16;

  const float* Hm  = H  + (size_t)m * M_ * S_;
  const float* Ybm = Yt + (size_t)(b * M_ + m) * SP_ * WP_;

  v8f acc0 = {}, acc1 = {}, acc2 = {};

#pragma unroll
  for (int s0 = 0; s0 < 24; s0 += 4) {       // K over s, zero-pad past 22
    v2f a, bv0, bv1, bv2;
#pragma unroll
    for (int j = 0; j < 2; ++j) {
      int s = s0 + j + 2 * hi;
      bool sok = (s < S_);
      int sc = sok ? s : S_ - 1;
      // unconditional in-bounds loads, then select-to-zero
      float av = Hm[(n0 + lo16) * S_ + sc];
      const float* row = Ybm + (size_t)sc * WP_ + lo16;
      float v0 = row[(t + 0) * 16];
      float v1 = row[(t + 1) * 16];
      float v2 = row[(t + 2) * 16];
      a[j]   = sok ? av : 0.0f;
      bv0[j] = sok ? v0 : 0.0f;
      bv1[j] = sok ? v1 : 0.0f;
      bv2[j] = sok ? v2 : 0.0f;
    }
    acc0 = __builtin_amdgcn_wmma_f32_16x16x4_f32(false, a, false, bv0,
                                                 (short)0, acc0, false, false);
    acc1 = __builtin_amdgcn_wmma_f32_16x16x4_f32(false, a, false, bv1,
                                                 (short)0, acc1, false, false);
    acc2 = __builtin_amdgcn_wmma_f32_16x16x4_f32(false, a, false, bv2,
                                                 (short)0, acc2, false, false);
  }

  // Hsum[n0+j] held by lane j (j<16); broadcast the 8 rows this lane extracts
  float hsum = 0.0f;
  {
    const float* hr = Hm + (n0 + (lane & 15)) * S_;
    if (lane < 16) {
#pragma unroll
      for (int s = 0; s < S_; ++s) hsum += hr[s];
    }
  }
  float hs[8];
#pragma unroll
  for (int r = 0; r < 8; ++r) hs[r] = __shfl(hsum, r + 8 * hi, 32);

  const float noiseScale =
      sqrtf(*sumYY * (1.0f / ((float)M_ * (float)W_ * (float)B_ * 10000.0f)));
  const float* epsbm = eps + (size_t)(b * M_ + m) * W_;
  float* Xbm = Xout + (size_t)(b * M_ + m) * M_ * L_;

  float lmax = -3.4e38f;
#pragma unroll
  for (int ut = 0; ut < 3; ++ut) {
    const v8f& acc = (ut == 0) ? acc0 : (ut == 1) ? acc1 : acc2;
    int u = (t + ut) * 16 + lo16;            // D col (same for all rows)
    int uc = u < W_ ? u : W_ - 1;
    float ev = epsbm[uc];
    ev = (u < W_) ? ev : 0.0f;
    float nsev = noiseScale * ev;
#pragma unroll
    for (int r = 0; r < 8; ++r) {
      int n = n0 + r + 8 * hi;               // D row
      int l = u - n;
      bool ok = (unsigned)l < (unsigned)L_;  // implies u < 279
      float v = fmaf(nsev, hs[r], acc[r]);
      // branch-free store: invalid elements land in the dump word
      float* addr = ok ? (Xbm + (size_t)n * L_ + l) : dump;
      *addr = v;
      lmax = fmaxf(lmax, ok ? v : -3.4e38f);
    }
  }
#pragma unroll
  for (int off = 16; off > 0; off >>= 1)
    lmax = fmaxf(lmax, __shfl_down(lmax, off, 32));
  if (lane == 0) atomicMax(umax, map_f32(lmax));
}

// ---------------------------------------------------------------------------
// Phase 3: out /= max(X)
// ---------------------------------------------------------------------------
__global__ __launch_bounds__(256) void cassi_norm(
    float4* __restrict__ out, const unsigned* __restrict__ umax, int n4)
{
  int i = blockIdx.x * blockDim.x + threadIdx.x;
  unsigned u = *umax;
  float mx = (u & 0x80000000u) ? __uint_as_float(u ^ 0x80000000u)
                               : __uint_as_float(~u);
  if (i < n4) {
    float4 v = out[i];
    v.x /= mx; v.y /= mx; v.z /= mx; v.w /= mx;
    out[i] = v;
  }
}

extern "C" void kernel_launch(void* const* d_in, const int* in_sizes, int n_in,
                              void* d_out, int out_size, void* d_ws, size_t ws_size,
                              hipStream_t stream) {
  const float* x   = (const float*)d_in[0];   // (B, M, M, L)
  const float* H   = (const float*)d_in[1];   // (1, M, M, 1, S)
  const float* eps = (const float*)d_in[2];   // (B, M, W, 1)
  float* out = (float*)d_out;                 // (B, M, M, L)

  float*    sumYY = (float*)d_ws;
  unsigned* umax  = (unsigned*)((char*)d_ws + 4);
  float*    dump  = (float*)((char*)d_ws + 16);
  float*    Yt    = (float*)((char*)d_ws + 256);   // [B][M][SP][WP] ~28 MB

  hipMemsetAsync(d_ws, 0, 8, stream);

  cassi_phase1<<<(B_ * M_ * NTW) / 4, 128, 0, stream>>>(x, H, Yt, sumYY);
  cassi_phase2<<<(B_ * M_ * NTN) / 4, 128, 0, stream>>>(H, eps, Yt, sumYY,
                                                        out, umax, dump);
  int n4 = (B_ * M_ * M_ * L_) / 4;
  cassi_norm<<<(n4 + 255) / 256, 256, 0, stream>>>((float4*)out, umax, n4);
}